// MambaBranch_72825465471239
// MI455X (gfx1250) — compile-verified
//
#include <hip/hip_runtime.h>
#include <math.h>

typedef __attribute__((ext_vector_type(2))) float v2f;
typedef __attribute__((ext_vector_type(8))) float v8f;

#define BB     8
#define LL     512
#define HIDD   256
#define LATD   256
#define INNERD 512
#define NLAYER 3
#define NTOK   (BB * LL)   // 4096 token rows
#define CH     8           // scan chunks per sequence
#define CT     (LL / CH)   // 64 timesteps per chunk

// ---------------------------------------------------------------------------
// WMMA f32 GEMM: C[M,N] = A[M,K] x B[K,N]. One wave computes a 32x64 tile:
// two 16-row A fragments share four B fragments -> 8 WMMAs per 10 loads.
// A fragment (16x4 f32): lane<16 -> row M=lane, {K+0,K+1}; lane>=16 -> {K+2,K+3}
// B fragment (4x16 f32): lane<16 -> col N=lane, rows {K+0,K+1}; lane>=16 -> {K+2,K+3}
// C/D (16x16 f32): VGPR v, lane<16 -> M=v, N=lane; lane>=16 -> M=v+8, N=lane-16
// ---------------------------------------------------------------------------
__global__ __launch_bounds__(32)
void gemm_wmma_f32_32x64(const float* __restrict__ A, const float* __restrict__ B,
                         float* __restrict__ C, int M, int N, int K) {
  const int lane  = threadIdx.x;     // 0..31, wave32
  const int half  = lane >> 4;       // 0 or 1
  const int lid   = lane & 15;
  const int mBase = blockIdx.y << 5;
  const int nBase = blockIdx.x << 6;

  const float* Arow0 = A + (size_t)(mBase + lid) * K + 2 * half;
  const float* Arow1 = Arow0 + (size_t)16 * K;

  v8f acc00 = {}, acc01 = {}, acc02 = {}, acc03 = {};
  v8f acc10 = {}, acc11 = {}, acc12 = {}, acc13 = {};

  for (int k = 0; k < K; k += 4) {
    v2f a0 = *(const v2f*)(Arow0 + k);
    v2f a1 = *(const v2f*)(Arow1 + k);
    const float* Bk  = B + (size_t)(k + 2 * half) * N + nBase + lid;
    const float* Bk1 = Bk + N;
    v2f b0 = { Bk[0],  Bk1[0]  };
    v2f b1 = { Bk[16], Bk1[16] };
    v2f b2 = { Bk[32], Bk1[32] };
    v2f b3 = { Bk[48], Bk1[48] };
    acc00 = __builtin_amdgcn_wmma_f32_16x16x4_f32(false, a0, false, b0, (short)0, acc00, false, false);
    acc10 = __builtin_amdgcn_wmma_f32_16x16x4_f32(false, a1, false, b0, (short)0, acc10, false, false);
    acc01 = __builtin_amdgcn_wmma_f32_16x16x4_f32(false, a0, false, b1, (short)0, acc01, false, false);
    acc11 = __builtin_amdgcn_wmma_f32_16x16x4_f32(false, a1, false, b1, (short)0, acc11, false, false);
    acc02 = __builtin_amdgcn_wmma_f32_16x16x4_f32(false, a0, false, b2, (short)0, acc02, false, false);
    acc12 = __builtin_amdgcn_wmma_f32_16x16x4_f32(false, a1, false, b2, (short)0, acc12, false, false);
    acc03 = __builtin_amdgcn_wmma_f32_16x16x4_f32(false, a0, false, b3, (short)0, acc03, false, false);
    acc13 = __builtin_amdgcn_wmma_f32_16x16x4_f32(false, a1, false, b3, (short)0, acc13, false, false);
  }

  float* Cr0 = C + (size_t)(mBase + 8 * half) * N + nBase + lid;
  float* Cr1 = Cr0 + (size_t)16 * N;
  #pragma unroll
  for (int v = 0; v < 8; ++v) {
    float* p0 = Cr0 + (size_t)v * N;
    float* p1 = Cr1 + (size_t)v * N;
    p0[0]  = acc00[v];  p0[16] = acc01[v];  p0[32] = acc02[v];  p0[48] = acc03[v];
    p1[0]  = acc10[v];  p1[16] = acc11[v];  p1[32] = acc12[v];  p1[48] = acc13[v];
  }
}

// ---------------------------------------------------------------------------
// Row LayerNorm over 256 columns. One 256-thread block per row.
// ---------------------------------------------------------------------------
__global__ __launch_bounds__(256)
void layernorm_kernel(const float* __restrict__ x, const float* __restrict__ g,
                      const float* __restrict__ b, float* __restrict__ out) {
  const int r = blockIdx.x, c = threadIdx.x;
  __shared__ float s1[256], s2[256];
  float v = x[(size_t)r * 256 + c];
  s1[c] = v; s2[c] = v * v;
  __syncthreads();
  for (int off = 128; off > 0; off >>= 1) {
    if (c < off) { s1[c] += s1[c + off]; s2[c] += s2[c + off]; }
    __syncthreads();
  }
  const float mu  = s1[0] * (1.0f / 256.0f);
  const float var = s2[0] * (1.0f / 256.0f) - mu * mu;
  const float inv = rsqrtf(var + 1e-5f);
  out[(size_t)r * 256 + c] = (v - mu) * inv * g[c] + b[c];
}

// ---------------------------------------------------------------------------
// RMSNorm + residual add: h[r,:] += rmsnorm(x[r,:], w)
// ---------------------------------------------------------------------------
__global__ __launch_bounds__(256)
void rmsnorm_add_kernel(const float* __restrict__ x, const float* __restrict__ w,
                        float* __restrict__ h) {
  const int r = blockIdx.x, c = threadIdx.x;
  __shared__ float s2[256];
  float v = x[(size_t)r * 256 + c];
  s2[c] = v * v;
  __syncthreads();
  for (int off = 128; off > 0; off >>= 1) {
    if (c < off) s2[c] += s2[c + off];
    __syncthreads();
  }
  const float inv = rsqrtf(s2[0] * (1.0f / 256.0f) + 1e-6f);
  h[(size_t)r * 256 + c] += v * inv * w[c];
}

// ---------------------------------------------------------------------------
// Chunked decay scan, pass 1: conv3 + chunk-local scan + end-carry.
// One thread per (batch, chunk, channel): B*CH*INNER = 32768 threads.
// sloc gets the chunk-local (zero-initial-state) scan; carry gets its end value.
// ---------------------------------------------------------------------------
__global__ __launch_bounds__(256)
void scan_part1(const float* __restrict__ xp, const float* __restrict__ conv_w,
                const float* __restrict__ decay, const float* __restrict__ scale,
                float* __restrict__ sloc, float* __restrict__ carry) {
  const int idx   = blockIdx.x * blockDim.x + threadIdx.x;
  const int c     = idx & (INNERD - 1);
  const int rest  = idx >> 9;
  const int chunk = rest & (CH - 1);
  const int b     = rest >> 3;
  const int t0    = chunk * CT;

  const float w0 = conv_w[c * 3 + 0];
  const float w1 = conv_w[c * 3 + 1];
  const float w2 = conv_w[c * 3 + 2];
  const float d    = 1.0f / (1.0f + expf(-decay[c]));
  const float dc   = fmaxf(d, 1e-6f);
  const float gain = (1.0f - d) * scale[c];

  const float* up = xp + (size_t)b * LL * (2 * INNERD) + c;
  float um1 = (t0 > 0) ? up[(size_t)(t0 - 1) * (2 * INNERD)] : 0.0f;
  float um2 = (t0 > 0) ? up[(size_t)(t0 - 2) * (2 * INNERD)] : 0.0f;

  const float* ut = up + (size_t)t0 * (2 * INNERD);
  float* sp = sloc + ((size_t)b * LL + t0) * INNERD + c;

  float acc = 0.0f;
  for (int t = 0; t < CT; ++t) {
    const float u  = ut[(size_t)t * (2 * INNERD)];
    const float uc = w0 * um2 + w1 * um1 + w2 * u;
    um2 = um1; um1 = u;
    acc = dc * acc + gain * uc;
    sp[(size_t)t * INNERD] = acc;
  }
  carry[((size_t)b * CH + chunk) * INNERD + c] = acc;
}

// ---------------------------------------------------------------------------
// Chunked decay scan, pass 2: fold carry prefix, apply correction + sigmoid gate.
// s_global[t0+tau] = sloc[t0+tau] + dc^(tau+1) * cin,  cin = s_global[t0-1].
// cin folded over previous chunks' carries with dT = dc^CT (6 squarings).
// y is updated in place (sloc buffer becomes gated y).
// ---------------------------------------------------------------------------
__global__ __launch_bounds__(256)
void scan_part2(const float* __restrict__ xp, const float* __restrict__ decay,
                const float* __restrict__ carry, float* __restrict__ y) {
  const int idx   = blockIdx.x * blockDim.x + threadIdx.x;
  const int c     = idx & (INNERD - 1);
  const int rest  = idx >> 9;
  const int chunk = rest & (CH - 1);
  const int b     = rest >> 3;
  const int t0    = chunk * CT;

  const float d  = 1.0f / (1.0f + expf(-decay[c]));
  const float dc = fmaxf(d, 1e-6f);
  float dT = dc;                       // dc^64 via 6 squarings
  dT *= dT; dT *= dT; dT *= dT; dT *= dT; dT *= dT; dT *= dT;

  const float* cp = carry + (size_t)b * CH * INNERD + c;
  float cin = 0.0f;
  for (int j = 0; j < chunk; ++j) cin = dT * cin + cp[(size_t)j * INNERD];

  const float* gt = xp + ((size_t)b * LL + t0) * (2 * INNERD) + INNERD + c;
  float* yp = y + ((size_t)b * LL + t0) * INNERD + c;

  float p = dc;
  for (int t = 0; t < CT; ++t) {
    const float s  = yp[(size_t)t * INNERD] + p * cin;
    p *= dc;
    const float gg = gt[(size_t)t * (2 * INNERD)];
    const float sg = 1.0f / (1.0f + expf(-gg));
    yp[(size_t)t * INNERD] = s * sg;
  }
}

// ---------------------------------------------------------------------------
// pooled[b,c] = 0.5*mean_t h[b,t,c] + 0.5*h[b,L-1,c]
// ---------------------------------------------------------------------------
__global__ __launch_bounds__(256)
void pool_kernel(const float* __restrict__ h, float* __restrict__ pooled) {
  const int b = blockIdx.x, c = threadIdx.x;
  const float* p = h + (size_t)b * LL * LATD + c;
  float s = 0.0f;
  for (int t = 0; t < LL; ++t) s += p[(size_t)t * LATD];
  pooled[b * LATD + c] = 0.5f * (s * (1.0f / (float)LL)) + 0.5f * p[(size_t)(LL - 1) * LATD];
}

// ---------------------------------------------------------------------------
// Head: z = GELU_exact(LayerNorm(pooled @ w1 + b1)); out = z @ w2 + b2
// ---------------------------------------------------------------------------
__global__ __launch_bounds__(256)
void head_kernel(const float* __restrict__ pooled,
                 const float* __restrict__ w1, const float* __restrict__ b1,
                 const float* __restrict__ lng, const float* __restrict__ lnb,
                 const float* __restrict__ w2, const float* __restrict__ b2,
                 float* __restrict__ out) {
  __shared__ float pr[256], zs[256], s1[256], s2[256];
  const int b = blockIdx.x, c = threadIdx.x;
  pr[c] = pooled[b * 256 + c];
  __syncthreads();
  float acc = b1[c];
  for (int k = 0; k < 256; ++k) acc += pr[k] * w1[k * 256 + c];
  s1[c] = acc; s2[c] = acc * acc;
  __syncthreads();
  for (int off = 128; off > 0; off >>= 1) {
    if (c < off) { s1[c] += s1[c + off]; s2[c] += s2[c + off]; }
    __syncthreads();
  }
  const float mu  = s1[0] * (1.0f / 256.0f);
  const float var = s2[0] * (1.0f / 256.0f) - mu * mu;
  const float zn  = (acc - mu) * rsqrtf(var + 1e-5f) * lng[c] + lnb[c];
  zs[c] = 0.5f * zn * (1.0f + erff(zn * 0.70710678118654752f));  // exact GELU
  __syncthreads();
  for (int j = c; j < 768; j += 256) {
    float a = b2[j];
    for (int k = 0; k < 256; ++k) a += zs[k] * w2[(size_t)k * 768 + j];
    out[(size_t)b * 768 + j] = a;
  }
}

// ---------------------------------------------------------------------------
extern "C" void kernel_launch(void* const* d_in, const int* in_sizes, int n_in,
                              void* d_out, int out_size, void* d_ws, size_t ws_size,
                              hipStream_t stream) {
  const float* x        = (const float*)d_in[0];
  const float* in_proj  = (const float*)d_in[1];
  const float* in_ln_g  = (const float*)d_in[2];
  const float* in_ln_b  = (const float*)d_in[3];
  const float* norms_g  = (const float*)d_in[4];
  const float* norms_b  = (const float*)d_in[5];
  const float* blk_in_w = (const float*)d_in[6];
  const float* blk_conv = (const float*)d_in[7];
  const float* blk_dec  = (const float*)d_in[8];
  const float* blk_scl  = (const float*)d_in[9];
  const float* blk_out  = (const float*)d_in[10];
  const float* blk_rms  = (const float*)d_in[11];
  const float* head_w1  = (const float*)d_in[12];
  const float* head_b1  = (const float*)d_in[13];
  const float* head_lg  = (const float*)d_in[14];
  const float* head_lb  = (const float*)d_in[15];
  const float* head_w2  = (const float*)d_in[16];
  const float* head_b2  = (const float*)d_in[17];

  float* ws    = (float*)d_ws;
  float* h     = ws;                 // [4096,256]   = 1,048,576 f
  float* tmp   = ws + 1048576;       // [4096,256]   (pre-norm GEMM out / hn)
  float* xp    = ws + 2097152;       // [4096,1024]  = 4,194,304 f
  float* y     = ws + 6291456;       // [4096,512]   = 2,097,152 f (sloc -> gated y)
  float* pool  = ws + 8388608;       // [8,256]      = 2,048 f
  float* carry = ws + 8390656;       // [8,8,512]    = 32,768 f
  float* out   = (float*)d_out;      // [8,768] -> reshaped [8,96,8] by harness

  const dim3 wave(32);
  const int scanBlocks = (BB * CH * INNERD) / 256;   // 128

  // stem: tmp = x @ in_proj ; h = LN(tmp)
  gemm_wmma_f32_32x64<<<dim3(LATD / 64, NTOK / 32), wave, 0, stream>>>(
      x, in_proj, tmp, NTOK, LATD, HIDD);
  layernorm_kernel<<<NTOK, 256, 0, stream>>>(tmp, in_ln_g, in_ln_b, h);

  for (int i = 0; i < NLAYER; ++i) {
    // hn = LN(h)
    layernorm_kernel<<<NTOK, 256, 0, stream>>>(h, norms_g + i * LATD, norms_b + i * LATD, tmp);
    // xp = hn @ blk_in_w[i]   [4096,256]x[256,1024]
    gemm_wmma_f32_32x64<<<dim3((2 * INNERD) / 64, NTOK / 32), wave, 0, stream>>>(
        tmp, blk_in_w + (size_t)i * LATD * 2 * INNERD, xp, NTOK, 2 * INNERD, LATD);
    // conv3 + chunked decay scan + sigmoid gate -> y
    scan_part1<<<scanBlocks, 256, 0, stream>>>(
        xp, blk_conv + (size_t)i * INNERD * 3, blk_dec + (size_t)i * INNERD,
        blk_scl + (size_t)i * INNERD, y, carry);
    scan_part2<<<scanBlocks, 256, 0, stream>>>(
        xp, blk_dec + (size_t)i * INNERD, carry, y);
    // tmp = y @ blk_out_w[i]  [4096,512]x[512,256]
    gemm_wmma_f32_32x64<<<dim3(LATD / 64, NTOK / 32), wave, 0, stream>>>(
        y, blk_out + (size_t)i * INNERD * LATD, tmp, NTOK, LATD, INNERD);
    // h += rmsnorm(tmp)
    rmsnorm_add_kernel<<<NTOK, 256, 0, stream>>>(tmp, blk_rms + (size_t)i * LATD, h);
  }

  pool_kernel<<<BB, 256, 0, stream>>>(h, pool);
  head_kernel<<<BB, 256, 0, stream>>>(pool, head_w1, head_b1, head_lg, head_lb,
                                      head_w2, head_b2, out);
}